// FourierAttention_73014444032546
// MI455X (gfx1250) — compile-verified
//
#include <hip/hip_runtime.h>
#include <hip/hip_bf16.h>
#include <math.h>

typedef float v2f __attribute__((ext_vector_type(2)));
typedef float v8f __attribute__((ext_vector_type(8)));

#define SEQ     4096
#define NCOL    16384     // 16 * 1024 columns, also the seq-axis stride in floats
#define THREADS 256       // 8 wave32s

// Array index n = n0 + 16*n1 + 256*n2.  Stage<AXIS> transforms along axis nAXIS,
// in place, over 256 "lines" of 16 elements each.
template<int AXIS>
__device__ __forceinline__ int posOf(int e, int L) {
  if constexpr (AXIS == 2)      return L + (e << 8);                           // stride 256
  else if constexpr (AXIS == 1) return (L & 15) + ((L >> 4) << 8) + (e << 4);  // stride 16
  else                          return (L << 4) + e;                           // stride 1
}

// One radix-16 complex DFT stage done entirely with V_WMMA_F32_16X16X4_F32.
// s = +1: forward table (e^{-2pi i mk/16});  s = -1: conjugate (inverse).
template<int AXIS>
__device__ __forceinline__ void dft16_stage(float* __restrict__ re, float* __restrict__ im,
                                            const float* __restrict__ w16r,
                                            const float* __restrict__ w16i,
                                            float s, int lane, int wave)
{
  const int n    = lane & 15;   // B/C/D column, and A row (M)
  const int half = lane >> 4;   // K-half select within each K=4 chunk
  #pragma unroll
  for (int jj = 0; jj < 2; ++jj) {
    const int j = wave + 8 * jj;       // tile index 0..15, uniform per wave
    const int L = 16 * j + n;          // this lane's line
    v8f cr = {0.f,0.f,0.f,0.f,0.f,0.f,0.f,0.f};
    v8f ci = {0.f,0.f,0.f,0.f,0.f,0.f,0.f,0.f};
    #pragma unroll
    for (int kk = 0; kk < 4; ++kk) {   // K = 16 as four K=4 WMMA steps
      const int kb = 4 * kk + 2 * half;
      const float wr0 = w16r[n * 16 + kb];
      const float wr1 = w16r[n * 16 + kb + 1];
      const float wi0 = s * w16i[n * 16 + kb];
      const float wi1 = s * w16i[n * 16 + kb + 1];
      v2f aR  = {  wr0,  wr1 };
      v2f aI  = {  wi0,  wi1 };
      v2f aIn = { -wi0, -wi1 };
      const int p0 = posOf<AXIS>(kb,     L);
      const int p1 = posOf<AXIS>(kb + 1, L);
      v2f bR = { re[p0], re[p1] };
      v2f bI = { im[p0], im[p1] };
      // Yr += Wr*Xr - Wi*Xi ; Yi += Wr*Xi + Wi*Xr
      cr = __builtin_amdgcn_wmma_f32_16x16x4_f32(false, aR,  false, bR, (short)0, cr, false, false);
      cr = __builtin_amdgcn_wmma_f32_16x16x4_f32(false, aIn, false, bI, (short)0, cr, false, false);
      ci = __builtin_amdgcn_wmma_f32_16x16x4_f32(false, aR,  false, bI, (short)0, ci, false, false);
      ci = __builtin_amdgcn_wmma_f32_16x16x4_f32(false, aI,  false, bR, (short)0, ci, false, false);
    }
    #pragma unroll
    for (int i = 0; i < 8; ++i) {      // D layout: lane n, rows i + 8*half
      const int m = i + 8 * half;
      const int p = posOf<AXIS>(m, L);
      re[p] = cr[i];
      im[p] = ci[i];
    }
  }
}

__global__ __launch_bounds__(THREADS)
void FourierAttention_73014444032546_kernel(const float* __restrict__ x,
                                            float* __restrict__ out)
{
  __shared__ float re[SEQ];
  __shared__ float im[SEQ];
  __shared__ float w16r[256], w16i[256], t1r[256], t1i[256];

  const int tid  = threadIdx.x;
  const int lane = tid & 31;
  const int wave = tid >> 5;
  const int b    = blockIdx.x;          // column in [0, 16384)

  // Twiddle tables: W16[m][k] = e^{-2pi i mk/16}; T1[m][k] = e^{-2pi i mk/256}
  {
    const int m = tid >> 4, k = tid & 15;
    float si, co;
    sincospif(-(float)(m * k) / 8.0f, &si, &co);      // -2*m*k/16 in pi units
    w16r[tid] = co; w16i[tid] = si;
    sincospif(-(float)(m * k) / 128.0f, &si, &co);    // -2*m*k/256 in pi units
    t1r[tid] = co; t1i[tid] = si;
  }

  // Load column b into LDS (seq stride 16384 floats); imag = 0.
  const float* xc = x + b;
  #pragma unroll
  for (int t = 0; t < SEQ / THREADS; ++t) {
    const int sidx = tid + t * THREADS;
    re[sidx] = xc[(size_t)sidx * NCOL];
    im[sidx] = 0.0f;
  }
  __syncthreads();

  // ---------------- forward FFT: A (stride 256), T1, B (stride 16), T2, C (stride 1)
  dft16_stage<2>(re, im, w16r, w16i, +1.0f, lane, wave);
  __syncthreads();
  for (int p = tid; p < SEQ; p += THREADS) {          // T1 = w256^{n1*k2}
    const int n1 = (p >> 4) & 15, k2 = p >> 8;
    const float tr = t1r[n1 * 16 + k2], ti = t1i[n1 * 16 + k2];
    const float r = re[p], q = im[p];
    re[p] = r * tr - q * ti;
    im[p] = r * ti + q * tr;
  }
  __syncthreads();
  dft16_stage<1>(re, im, w16r, w16i, +1.0f, lane, wave);
  __syncthreads();
  for (int p = tid; p < SEQ; p += THREADS) {          // T2 = w4096^{n0*(16k1+k2)}
    const int n0 = p & 15, k1 = (p >> 4) & 15, k2 = p >> 8;
    const int t = 16 * k1 + k2;
    float si, co;
    sincospif(-(float)(n0 * t) / 2048.0f, &si, &co);
    const float r = re[p], q = im[p];
    re[p] = r * co - q * si;
    im[p] = r * si + q * co;
  }
  __syncthreads();
  dft16_stage<0>(re, im, w16r, w16i, +1.0f, lane, wave);
  __syncthreads();

  // ---------------- inverse FFT: exact conjugate-transpose of each stage, reversed
  dft16_stage<0>(re, im, w16r, w16i, -1.0f, lane, wave);
  __syncthreads();
  for (int p = tid; p < SEQ; p += THREADS) {          // conj T2
    const int n0 = p & 15, k1 = (p >> 4) & 15, k2 = p >> 8;
    const int t = 16 * k1 + k2;
    float si, co;
    sincospif((float)(n0 * t) / 2048.0f, &si, &co);
    const float r = re[p], q = im[p];
    re[p] = r * co - q * si;
    im[p] = r * si + q * co;
  }
  __syncthreads();
  dft16_stage<1>(re, im, w16r, w16i, -1.0f, lane, wave);
  __syncthreads();
  for (int p = tid; p < SEQ; p += THREADS) {          // conj T1
    const int n1 = (p >> 4) & 15, k2 = p >> 8;
    const float tr = t1r[n1 * 16 + k2], ti = -t1i[n1 * 16 + k2];
    const float r = re[p], q = im[p];
    re[p] = r * tr - q * ti;
    im[p] = r * ti + q * tr;
  }
  __syncthreads();
  dft16_stage<2>(re, im, w16r, w16i, -1.0f, lane, wave);
  __syncthreads();

  // irfft returns the real part; scale by 1/N (each W16*conj(W16) pair contributes 16).
  float* oc = out + b;
  #pragma unroll
  for (int t = 0; t < SEQ / THREADS; ++t) {
    const int sidx = tid + t * THREADS;
    oc[(size_t)sidx * NCOL] = re[sidx] * (1.0f / 4096.0f);
  }
}

extern "C" void kernel_launch(void* const* d_in, const int* in_sizes, int n_in,
                              void* d_out, int out_size, void* d_ws, size_t ws_size,
                              hipStream_t stream) {
  (void)in_sizes; (void)n_in; (void)d_ws; (void)ws_size; (void)out_size;
  const float* x = (const float*)d_in[0];
  float* out = (float*)d_out;
  hipLaunchKernelGGL(FourierAttention_73014444032546_kernel,
                     dim3(NCOL), dim3(THREADS), 0, stream, x, out);
}